// MaxSigLayer_73469710565961
// MI455X (gfx1250) — compile-verified
//
#include <hip/hip_runtime.h>
#include <hip/hip_bf16.h>
#include <stdint.h>

#define TS     16                 // spatial tile
#define LW     18                 // tile + 2 halo
#define LN     (LW * LW)          // 324 raw entries
#define SSTR   20                 // padded row stride for sigmoid tile
#define Himg   128
#define Wimg   128
#define Cin    64
#define CHOUT  64
#define HW     (Himg * Wimg)

__device__ __forceinline__ float fsig(float v) {
    // sigmoid via fast exp + fast rcp (v_exp_f32 / v_rcp_f32)
    return __builtin_amdgcn_rcpf(1.0f + __expf(-v));
}

// gfx1250 async global -> LDS copy (ASYNCcnt path). VDST = LDS byte address,
// VADDR = 64-bit global address. Low 32 bits of a generic LDS pointer are the
// wave-relative LDS byte address (flat-aperture layout), matching the async
// LDS address operand (an offset added to the workgroup's LDS base).
__device__ __forceinline__ void async_ld_f32(unsigned ldsaddr, const float* g) {
    asm volatile("global_load_async_to_lds_b32 %0, %1, off"
                 :: "v"(ldsaddr), "v"(g)
                 : "memory");
}

#define ASYNC_WAIT(N) asm volatile("s_wait_asynccnt %0" :: "i"(N) : "memory")

__global__ __launch_bounds__(256, 2)
void maxsig_fused(const float* __restrict__ x,
                  const float* __restrict__ weight,
                  const float* __restrict__ bias,
                  const float* __restrict__ wc_p,
                  const float* __restrict__ wm_p,
                  float* __restrict__ out)
{
    __shared__ __align__(16) float rawA[LN + 4];   // +4: dummy slot for inactive lanes
    __shared__ __align__(16) float rawB[LN + 4];
    __shared__ __align__(16) float sig[LW * SSTR];

    const int tid = threadIdx.x;
    const int lx  = tid & 15;
    const int ly  = tid >> 4;
    const int w0  = blockIdx.x * TS;
    const int h0  = blockIdx.y * TS;
    const int b   = blockIdx.z;

    // per-tap sigmoid(weight), sigmoid(bias); uniform across channels
    float wk[9], bk[9];
#pragma unroll
    for (int i = 0; i < 9; ++i) {
        wk[i] = fsig(weight[i]);
        bk[i] = fsig(bias[i]);
    }
    const float wc = wc_p[0];
    const float wm = wm_p[0];

    // ---- load-entry assignment: entry0 = tid, entry1 = tid + 256 (if < 324) ----
    const int e0  = tid;
    const int ey0 = e0 / LW, ex0 = e0 % LW;
    int gh0 = h0 - 1 + ey0, gw0 = w0 - 1 + ex0;
    const bool oob0 = (gh0 < 0) | (gh0 >= Himg) | (gw0 < 0) | (gw0 >= Wimg);
    gh0 = min(max(gh0, 0), Himg - 1);
    gw0 = min(max(gw0, 0), Wimg - 1);
    const int off0  = gh0 * Wimg + gw0;
    const int sidx0 = ey0 * SSTR + ex0;

    const int e1  = tid + 256;
    const bool v1 = (e1 < LN);
    const int ey1 = e1 / LW, ex1 = e1 % LW;
    int gh1 = h0 - 1 + ey1, gw1 = w0 - 1 + ex1;
    const bool oob1 = (gh1 < 0) | (gh1 >= Himg) | (gw1 < 0) | (gw1 >= Wimg);
    gh1 = min(max(gh1, 0), Himg - 1);
    gw1 = min(max(gw1, 0), Wimg - 1);
    const int off1  = v1 ? (gh1 * Wimg + gw1) : 0;   // safe address for inactive lanes
    const int sidx1 = ey1 * SSTR + ex1;

    const unsigned lbA = (unsigned)(uintptr_t)(void*)rawA;
    const unsigned lbB = (unsigned)(uintptr_t)(void*)rawB;
    const unsigned d0  = (unsigned)(e0 * 4);
    const unsigned d1  = v1 ? (unsigned)(e1 * 4) : (unsigned)(LN * 4); // dummy slot

    const float* xb = x + (size_t)b * Cin * HW;

    // prologue: stream channel 0 tile into rawA (every lane issues 2 loads ->
    // ASYNCcnt is uniform: +2 per wave per channel)
    async_ld_f32(lbA + d0, xb + off0);
    async_ld_f32(lbA + d1, xb + off1);

    float acc = 0.0f;

    for (int c = 0; c < Cin; ++c) {
        const float*   cur = (c & 1) ? rawB : rawA;
        const unsigned lbn = (c & 1) ? lbA : lbB;   // buffer for channel c+1
        if (c + 1 < Cin) {
            const float* cb = xb + (size_t)(c + 1) * HW;
            async_ld_f32(lbn + d0, cb + off0);
            async_ld_f32(lbn + d1, cb + off1);
            ASYNC_WAIT(2);   // oldest 2 (channel c) complete; c+1 stays in flight
        } else {
            ASYNC_WAIT(0);
        }
        __syncthreads();     // all waves' async loads for channel c visible

        // sigmoid pass: raw -> sig, zero outside image (reference zero-pads sigmoid(x))
        {
            float s0v = oob0 ? 0.0f : fsig(cur[e0]);
            sig[sidx0] = s0v;
            if (v1) {
                float s1v = oob1 ? 0.0f : fsig(cur[e1]);
                sig[sidx1] = s1v;
            }
        }
        __syncthreads();

        // stencil pass: one pixel per thread
        const float* sp = &sig[ly * SSTR + lx];
        float p0 = sp[0],          p1 = sp[1],            p2 = sp[2];
        float p3 = sp[SSTR],       p4 = sp[SSTR + 1],     p5 = sp[SSTR + 2];
        float p6 = sp[2 * SSTR],   p7 = sp[2 * SSTR + 1], p8 = sp[2 * SSTR + 2];

        float D0 = fmaxf(wk[0], p0) + bk[0];
        float D1 = fmaxf(wk[1], p1) + bk[1];
        float D2 = fmaxf(wk[2], p2) + bk[2];
        float D3 = fmaxf(wk[3], p3) + bk[3];
        float D4 = fmaxf(wk[4], p4) + bk[4];
        float D5 = fmaxf(wk[5], p5) + bk[5];
        float D6 = fmaxf(wk[6], p6) + bk[6];
        float D7 = fmaxf(wk[7], p7) + bk[7];
        float D8 = fmaxf(wk[8], p8) + bk[8];

        float sum = ((D0 + D1) + (D2 + D3)) + ((D4 + D5) + (D6 + D7)) + D8;

        // exact median-of-9 via v_min3/v_med3/v_max3:
        // med3( max3(row mins), med3(row meds), min3(row maxes) )
        float lo0 = fminf(fminf(D0, D1), D2);
        float mi0 = __builtin_amdgcn_fmed3f(D0, D1, D2);
        float hi0 = fmaxf(fmaxf(D0, D1), D2);
        float lo1 = fminf(fminf(D3, D4), D5);
        float mi1 = __builtin_amdgcn_fmed3f(D3, D4, D5);
        float hi1 = fmaxf(fmaxf(D3, D4), D5);
        float lo2 = fminf(fminf(D6, D7), D8);
        float mi2 = __builtin_amdgcn_fmed3f(D6, D7, D8);
        float hi2 = fmaxf(fmaxf(D6, D7), D8);
        float mA  = fmaxf(fmaxf(lo0, lo1), lo2);
        float mB  = __builtin_amdgcn_fmed3f(mi0, mi1, mi2);
        float mC  = fminf(fminf(hi0, hi1), hi2);
        float med = __builtin_amdgcn_fmed3f(mA, mB, mC);

        // out_c = wc*center + wm*median - sum - sum/9 ; accumulate channel sum
        acc += wc * p4 + wm * med - sum - sum * (1.0f / 9.0f);
        __syncthreads();     // sig / raw buffers reused next iteration
    }

    // ---- broadcast channel-sum to all 64 output planes (b128 stores) ----
    rawA[tid] = acc;          // rawA free: last consumed at c=62, barrier passed
    __syncthreads();

    const int q  = tid & 63;          // float4 slot within tile
    const int g  = tid >> 6;          // plane group (4 groups x 16 planes)
    const int r  = q >> 2;            // tile row
    const int c4 = q & 3;             // float4 column
    const float4 vv = *(const float4*)(&rawA[(r << 4) + (c4 << 2)]);
    float* ob = out + ((size_t)b * CHOUT + (size_t)(g << 4)) * HW
                    + (size_t)(h0 + r) * Wimg + (size_t)(w0 + (c4 << 2));
#pragma unroll
    for (int k = 0; k < 16; ++k) {
        *(float4*)ob = vv;
        ob += HW;
    }
}

extern "C" void kernel_launch(void* const* d_in, const int* in_sizes, int n_in,
                              void* d_out, int out_size, void* d_ws, size_t ws_size,
                              hipStream_t stream) {
    const float* x  = (const float*)d_in[0];
    const float* wt = (const float*)d_in[1];
    const float* bs = (const float*)d_in[2];
    const float* wc = (const float*)d_in[3];
    const float* wm = (const float*)d_in[4];
    float* out = (float*)d_out;

    const int B = in_sizes[0] / (Cin * HW);   // 8
    dim3 grid(Wimg / TS, Himg / TS, B);
    dim3 block(256);
    hipLaunchKernelGGL(maxsig_fused, grid, block, 0, stream, x, wt, bs, wc, wm, out);
}